// AttnPool_2302102471544
// MI455X (gfx1250) — compile-verified
//
#include <hip/hip_runtime.h>
#include <hip/hip_bf16.h>

// ---------------- problem constants (match reference) ----------------
constexpr int DIM   = 128;     // feature dim
constexpr int NSEG  = 16384;   // number of segments S
constexpr int CHUNK = 256;     // rows per block in pooling kernel

typedef __attribute__((ext_vector_type(2))) float v2f;
typedef __attribute__((ext_vector_type(8))) float v8f;

// ---- monotonic uint encoding of float for atomicMax-based segment max ----
__device__ __forceinline__ unsigned f2ord(float f) {
  unsigned b = __float_as_uint(f);
  return b ^ ((unsigned)((int)b >> 31) | 0x80000000u);
}
__device__ __forceinline__ float ord2f(unsigned u) {
  unsigned b = (u & 0x80000000u) ? (u ^ 0x80000000u) : ~u;
  return __uint_as_float(b);
}
// f2ord(-FLT_MAX) == 0x00800000
#define ORD_NEG_MAX 0x00800000u

// ---------------- K0: zero output, init segment arrays ----------------
__global__ void attnpool_init(float* __restrict__ out,
                              unsigned* __restrict__ segmax,
                              float* __restrict__ denom,
                              int total_out) {
  int i = blockIdx.x * blockDim.x + threadIdx.x;
  if (i < total_out) out[i] = 0.0f;
  if (i < NSEG) { segmax[i] = ORD_NEG_MAX; denom[i] = 0.0f; }
}

// ---------------- K1: logits = h @ q_w + b ; segment max ----------------
// one wave32 per row; lane holds 4 of the 128 q_w values; b128 loads of h.
__global__ __launch_bounds__(256) void attnpool_logits(
    const float* __restrict__ h, const long long* __restrict__ batch,
    const float* __restrict__ qw, const float* __restrict__ qb,
    float* __restrict__ logits, unsigned* __restrict__ segmax, int n) {
  const int lane     = threadIdx.x & 31;
  const int waveId   = (blockIdx.x * blockDim.x + threadIdx.x) >> 5;
  const int numWaves = (gridDim.x * blockDim.x) >> 5;
  const float4 qv = *(const float4*)(qw + lane * 4);
  const float  qbv = qb[0];

  for (int row = waveId; row < n; row += numWaves) {
    const float4 hv = *(const float4*)(h + (long)row * DIM + lane * 4);
    if (row + numWaves < n)  // CDNA5 global_prefetch_b8 for next row
      __builtin_prefetch(h + (long)(row + numWaves) * DIM + lane * 4, 0, 0);
    float dot = hv.x * qv.x + hv.y * qv.y + hv.z * qv.z + hv.w * qv.w;
    #pragma unroll
    for (int off = 16; off > 0; off >>= 1)
      dot += __shfl_xor(dot, off, 32);
    if (lane == 0) {
      float lg = dot + qbv;
      logits[row] = lg;
      atomicMax(&segmax[(int)batch[row]], f2ord(lg));
    }
  }
}

// ---------------- K2: ex = exp(logit - segmax); denom += ex ----------------
__global__ void attnpool_expdenom(const long long* __restrict__ batch,
                                  float* __restrict__ lg_ex,       // in: logits, out: ex
                                  const unsigned* __restrict__ segmax,
                                  float* __restrict__ denom, int n) {
  int i = blockIdx.x * blockDim.x + threadIdx.x;
  if (i >= n) return;
  int sg = (int)batch[i];
  float e = expf(lg_ex[i] - ord2f(segmax[sg]));
  lg_ex[i] = e;
  atomicAdd(&denom[sg], e);
}

// ---------------- K3: pooled[s] += w_n * h[n]  via V_WMMA_F32_16X16X4_F32 ----
// Block = 256 threads = 8 waves; each wave owns a 16-column slice of DIM=128.
// Chunk of 256 sorted rows; 16 groups of 16 rows each. A sorted 16-row group
// has <= 16 distinct segments -> one 16x16 D tile per group:
//   A[m][k] = w_row if local_seg(row)==m else 0    (16x4 fp32)
//   B[k][n] = h[row][col_n]                        (4x16 fp32)
// D accumulates over 4 K-steps; tile flushed with atomicAdd per distinct seg.

// per-wave compute over one chunk. GUARD=true only for the final partial block.
template <bool GUARD>
__device__ __forceinline__ void pool_compute(
    const float* __restrict__ hp,   // h + (long)base*DIM + col
    int base, int n, int hf, int ln, int col,
    float* __restrict__ out,
    const float2* s_wl,             // .x = weight, .y = as_float(local idx)
    const int (*s_glist)[16], const int* s_nd) {
  const int k0 = hf ? 2 : 0;        // K slots owned by this half (VGPR0)
  const int k1 = hf ? 3 : 1;        // (VGPR1)

  for (int g = 0; g < 16; ++g) {
    const int rb = g * 16;
    const float* gp = hp + (long)rb * DIM;
    v8f c = {0.f, 0.f, 0.f, 0.f, 0.f, 0.f, 0.f, 0.f};
    #pragma unroll
    for (int k = 0; k < 16; k += 4) {
      const int rA0 = rb + k + k0;
      const int rA1 = rb + k + k1;
      // unconditional LDS b64 loads; select happens in VALU (v_cndmask)
      const float2 p0 = s_wl[rA0];
      const float2 p1 = s_wl[rA1];
      const int l0 = __float_as_int(p0.y);
      const int l1 = __float_as_int(p1.y);
      v2f a, b;
      a.x = (l0 == ln) ? p0.x : 0.0f;
      a.y = (l1 == ln) ? p1.x : 0.0f;
      if (GUARD) {
        b.x = (base + rA0 < n) ? gp[(long)(k + k0) * DIM] : 0.0f;
        b.y = (base + rA1 < n) ? gp[(long)(k + k1) * DIM] : 0.0f;
      } else {
        // constant byte offsets (k+kx)*512 -> immediate-offset global_load_b32
        b.x = gp[(k + k0) * DIM];
        b.y = gp[(k + k1) * DIM];
      }
      c = __builtin_amdgcn_wmma_f32_16x16x4_f32(
              false, a, false, b, (short)0, c, false, false);
    }
    // flush tile: D VGPR v holds M = v + 8*hf, N = lane%16
    const int nd = s_nd[g];
    #pragma unroll
    for (int v = 0; v < 8; ++v) {
      int m = hf * 8 + v;
      if (m < nd)
        atomicAdd(&out[(long)s_glist[g][m] * DIM + col], c[v]);
    }
  }
}

__global__ __launch_bounds__(256) void attnpool_wmma_pool(
    const float* __restrict__ h, const long long* __restrict__ batch,
    const float* __restrict__ ex, const float* __restrict__ denom,
    float* __restrict__ out, int n) {
  __shared__ float2 s_wl[CHUNK];     // .x = softmax weight, .y = local seg idx
  __shared__ int    s_seg[CHUNK];    // global segment id per row
  __shared__ int    s_glist[16][16]; // per group: distinct segment ids
  __shared__ int    s_nd[16];        // per group: number of distinct segments

  const int tid  = threadIdx.x;
  const int base = blockIdx.x * CHUNK;

  // ---- stage weights + segment ids (1 row per thread) ----
  {
    int gr = base + tid;
    if (gr < n) {
      int sg = (int)batch[gr];
      s_seg[tid]   = sg;
      s_wl[tid].x  = ex[gr] / denom[sg];
    } else {
      s_seg[tid]   = NSEG - 1;   // padded rows: weight 0, harmless segment id
      s_wl[tid].x  = 0.0f;
    }
  }
  __syncthreads();

  // ---- per-group distinct-segment scan (16 groups, one thread each) ----
  if (tid < 16) {
    int g = tid;
    int prev = s_seg[g * 16];
    int cnt  = 0;
    s_glist[g][0]     = prev;
    s_wl[g * 16].y    = __int_as_float(0);
    for (int i = 1; i < 16; ++i) {
      int sg = s_seg[g * 16 + i];
      if (sg != prev) { ++cnt; s_glist[g][cnt] = sg; prev = sg; }
      s_wl[g * 16 + i].y = __int_as_float(cnt);
    }
    s_nd[g] = cnt + 1;
  }
  __syncthreads();

  const int wave = tid >> 5;     // column slice 0..7
  const int lane = tid & 31;
  const int hf   = lane >> 4;    // half of the wave (K split / M split)
  const int ln   = lane & 15;    // A: M row ; B/D: column within slice
  const int col  = wave * 16 + ln;
  const float* hp = h + (long)base * DIM + col;

  if (base + CHUNK <= n) {       // block-uniform: all but the last block
    pool_compute<false>(hp, base, n, hf, ln, col, out, s_wl, s_glist, s_nd);
  } else {
    pool_compute<true >(hp, base, n, hf, ln, col, out, s_wl, s_glist, s_nd);
  }
}

// ---------------- launch ----------------
extern "C" void kernel_launch(void* const* d_in, const int* in_sizes, int n_in,
                              void* d_out, int out_size, void* d_ws, size_t ws_size,
                              hipStream_t stream) {
  const float*     h     = (const float*)d_in[0];
  const long long* batch = (const long long*)d_in[1];
  const float*     q_w   = (const float*)d_in[2];
  const float*     q_b   = (const float*)d_in[3];
  const int n = in_sizes[1];           // N rows
  float* out = (float*)d_out;

  // workspace layout: logits/ex [N] | segmax [S] | denom [S]  (~4.13 MB)
  float*    logits = (float*)d_ws;
  unsigned* segmax = (unsigned*)(logits + n);
  float*    denom  = (float*)(segmax + NSEG);

  const int total_out = NSEG * DIM;
  attnpool_init<<<(total_out + 255) / 256, 256, 0, stream>>>(out, segmax, denom, total_out);
  attnpool_logits<<<2048, 256, 0, stream>>>(h, batch, q_w, q_b, logits, segmax, n);
  attnpool_expdenom<<<(n + 255) / 256, 256, 0, stream>>>(batch, logits, segmax, denom, n);
  attnpool_wmma_pool<<<(n + CHUNK - 1) / CHUNK, 256, 0, stream>>>(h, batch, logits, denom, out, n);
}